// SparseLinear_88218628260171
// MI455X (gfx1250) — compile-verified
//
#include <hip/hip_runtime.h>
#include <cstdint>

// Problem constants (match reference)
#define OUTF   4096
#define INF    16384
#define BATCH  128
#define NNZ    2000000

// Tiling
#define NSLICES   32                      // nnz slices -> 32*8 = 256 workgroups
#define SLICE_LEN ((NNZ + NSLICES - 1) / NSLICES)   // 62500
#define CHUNK     1024                    // nnz per staged chunk
#define BCHUNK    16                      // batch elems per workgroup
#define ACCSTRIDE 17                      // 4096x17 fp32 accumulator, bank-conflict-free
#define THREADS   512                     // 16 wave32 waves

// CDNA5 async global->LDS copy, 8 bytes per lane, tracked by ASYNCcnt.
// NT near / RT far: don't pollute WGP$ (single use per WGP), retain in L2
// (COO stream is re-read by the 8 batch-chunk workgroups; 24 MB << 192 MB L2).
__device__ __forceinline__ void async_copy_b64(uint32_t lds_off, const void* gptr) {
    asm volatile("global_load_async_to_lds_b64 %0, %1, off th:TH_LOAD_NT_RT"
                 :: "v"(lds_off), "v"((uint64_t)(uintptr_t)gptr)
                 : "memory");
}

__global__ __launch_bounds__(THREADS, 1)
void spmm_scatter_kernel(const float* __restrict__ X,      // xT [INF][BATCH] (or raw inputs)
                         const float* __restrict__ vals,
                         const int*   __restrict__ rows,
                         const int*   __restrict__ cols,
                         float*       __restrict__ out,
                         int transposed) {
    extern __shared__ char smem[];
    int*   srows = (int*)  (smem);                    // [2][CHUNK]  8 KB
    int*   scols = (int*)  (smem + 2 * CHUNK * 4);    // [2][CHUNK]  8 KB
    float* svals = (float*)(smem + 4 * CHUNK * 4);    // [2][CHUNK]  8 KB
    float* acc   = (float*)(smem + 6 * CHUNK * 4);    // [OUTF][ACCSTRIDE]  272 KB

    const int tid         = threadIdx.x;
    const int slice_start = blockIdx.x * SLICE_LEN;
    const int slice_len   = min(SLICE_LEN, NNZ - slice_start);
    const int bchunk      = blockIdx.y;

    // Kick off async staging of the first chunk (uniform control flow, EXEC all-ones)
    {
        int gi = slice_start + 2 * tid;
        if (gi > NNZ - 2) gi = NNZ - 2;               // clamp tail (re-read, compute ignores)
        async_copy_b64((uint32_t)(uintptr_t)&srows[2 * tid], rows + gi);
        async_copy_b64((uint32_t)(uintptr_t)&scols[2 * tid], cols + gi);
        async_copy_b64((uint32_t)(uintptr_t)&svals[2 * tid], vals + gi);
    }

    // Zero the accumulator tile while the first loads are in flight
    for (int i = tid; i < OUTF * ACCSTRIDE; i += THREADS) acc[i] = 0.0f;

    const int  lane_b  = tid & (BCHUNK - 1);
    const int  bglob   = bchunk * BCHUNK + lane_b;
    const long cstride = transposed ? (long)BATCH : 1L;
    const float* xbase = transposed ? (X + bglob) : (X + (long)bglob * INF);

    int buf = 0;
    for (int base = 0; base < slice_len; base += CHUNK) {
        int nextbase = base + CHUNK;
        if (nextbase < slice_len) {
            // Prefetch next chunk into the other buffer, then wait for current
            // (per-wave ASYNCcnt, async loads complete in order: <=3 => current done)
            int gi = slice_start + nextbase + 2 * tid;
            if (gi > NNZ - 2) gi = NNZ - 2;
            int nb = buf ^ 1;
            async_copy_b64((uint32_t)(uintptr_t)&srows[nb * CHUNK + 2 * tid], rows + gi);
            async_copy_b64((uint32_t)(uintptr_t)&scols[nb * CHUNK + 2 * tid], cols + gi);
            async_copy_b64((uint32_t)(uintptr_t)&svals[nb * CHUNK + 2 * tid], vals + gi);
            asm volatile("s_wait_asynccnt 0x3" ::: "memory");
        } else {
            asm volatile("s_wait_asynccnt 0x0" ::: "memory");
        }
        __syncthreads();

        const int nvalid = min(CHUNK, slice_len - base);
        // 16 lanes per nnz (one per batch element of this chunk); LDS reads broadcast.
        // Unroll so the scheduler keeps several independent L2 gathers in flight.
#pragma unroll 4
        for (int j = tid >> 4; j < nvalid; j += THREADS / BCHUNK) {
            int   r = srows[buf * CHUNK + j];
            int   c = scols[buf * CHUNK + j];
            float v = svals[buf * CHUNK + j];
            float x = xbase[(long)c * cstride];                 // coalesced 64B per group
            atomicAdd(&acc[r * ACCSTRIDE + lane_b], v * x);     // ds_add_f32 (no rtn)
        }
        __syncthreads();   // protect staging buffer reuse
        buf ^= 1;
    }

    // Flush: coalesced global_atomic_add_f32 onto bias-initialized output.
    // LDS read stride 17 dwords across lanes -> all 64 banks hit, conflict-free.
    for (int i = tid; i < OUTF * BCHUNK; i += THREADS) {
        int bl = i >> 12;                 // batch lane 0..15
        int r  = i & (OUTF - 1);          // consecutive r across consecutive lanes
        float v = acc[r * ACCSTRIDE + bl];
        unsafeAtomicAdd(&out[(size_t)(bchunk * BCHUNK + bl) * OUTF + r], v);
    }
}

// out[b,r] = bias[r]  (also re-initializes output on every graph replay)
__global__ void bias_init_kernel(const float* __restrict__ bias, float* __restrict__ out) {
    int i = blockIdx.x * blockDim.x + threadIdx.x;
    if (i < BATCH * OUTF) out[i] = bias[i & (OUTF - 1)];
}

// xT[c][b] = inputs[b][c]  (32x32 LDS tiles, padded)
__global__ void transpose_kernel(const float* __restrict__ in, float* __restrict__ out) {
    __shared__ float tile[32][33];
    int tx = threadIdx.x, ty = threadIdx.y;     // 32 x 8
    int col0 = blockIdx.x * 32;
    int b0   = blockIdx.y * 32;
#pragma unroll
    for (int i = 0; i < 32; i += 8)
        tile[ty + i][tx] = in[(size_t)(b0 + ty + i) * INF + col0 + tx];
    __syncthreads();
#pragma unroll
    for (int i = 0; i < 32; i += 8)
        out[(size_t)(col0 + ty + i) * BATCH + b0 + tx] = tile[tx][ty + i];
}

extern "C" void kernel_launch(void* const* d_in, const int* in_sizes, int n_in,
                              void* d_out, int out_size, void* d_ws, size_t ws_size,
                              hipStream_t stream) {
    const float* inp  = (const float*)d_in[0];   // [128, 16384] f32
    const float* vals = (const float*)d_in[1];   // [2M] f32
    const float* bias = (const float*)d_in[2];   // [4096] f32
    const int*   rows = (const int*)  d_in[3];   // [2M] i32
    const int*   cols = (const int*)  d_in[4];   // [2M] i32
    float* out = (float*)d_out;
    float* xT  = (float*)d_ws;

    const size_t xt_bytes = (size_t)INF * BATCH * sizeof(float);   // 8 MB
    const int transposed = (d_ws != nullptr && ws_size >= xt_bytes) ? 1 : 0;

    bias_init_kernel<<<(BATCH * OUTF + 255) / 256, 256, 0, stream>>>(bias, out);

    if (transposed)
        transpose_kernel<<<dim3(INF / 32, BATCH / 32), dim3(32, 8), 0, stream>>>(inp, xT);

    const size_t shmem = 6 * CHUNK * 4 + (size_t)OUTF * ACCSTRIDE * 4;  // 303,104 B < 320 KB
    spmm_scatter_kernel<<<dim3(NSLICES, BATCH / BCHUNK), THREADS, shmem, stream>>>(
        transposed ? xT : inp, vals, rows, cols, out, transposed);
}